// GraphEncoder_65317862637644
// MI455X (gfx1250) — compile-verified
//
#include <hip/hip_runtime.h>
#include <hip/hip_bf16.h>

#define N_NODES  100000
#define N_EDGES  3200000
#define N_GRAPHS 128
#define IN_DIM   5
#define HID      64
#define XCOLS    6

typedef __attribute__((ext_vector_type(2))) float v2f;
typedef __attribute__((ext_vector_type(8))) float v8f;

// ---------------- degree / normalization ----------------
__global__ void k_init_deg(unsigned int* deg) {
    int n = blockIdx.x * blockDim.x + threadIdx.x;
    if (n < N_NODES) deg[n] = 1u;             // self loop
}

__global__ void k_count_deg(const long long* __restrict__ dst, unsigned int* deg) {
    int e = blockIdx.x * blockDim.x + threadIdx.x;
    if (e < N_EDGES) atomicAdd(&deg[(int)dst[e]], 1u);
}

__global__ void k_dis(const unsigned int* __restrict__ deg, float* __restrict__ dis) {
    int n = blockIdx.x * blockDim.x + threadIdx.x;
    if (n < N_NODES) dis[n] = rsqrtf((float)deg[n]);   // deg >= 1 always
}

// ---------------- layer 1: propagate 5-dim raw features ----------------
// agg1[n] = dis[n]^2 * feat[n]  (self loop term)
__global__ void k_self1(const float* __restrict__ x, const float* __restrict__ dis,
                        float* __restrict__ agg1) {
    int t = blockIdx.x * blockDim.x + threadIdx.x;
    if (t < N_NODES * IN_DIM) {
        int n = t / IN_DIM;
        int j = t - n * IN_DIM;
        float d = dis[n];
        agg1[t] = x[(long)n * XCOLS + 1 + j] * d * d;
    }
}

// agg1[d] += dis[s]*dis[d] * feat[s]   (one thread per edge, 5 atomics)
__global__ void k_edge1(const long long* __restrict__ src, const long long* __restrict__ dst,
                        const float* __restrict__ x, const float* __restrict__ dis,
                        float* __restrict__ agg1) {
    int e = blockIdx.x * blockDim.x + threadIdx.x;
    if (e < N_EDGES) {
        int s = (int)src[e], d = (int)dst[e];
        float w = dis[s] * dis[d];
#pragma unroll
        for (int j = 0; j < IN_DIM; ++j)
            atomicAdd(&agg1[(long)d * IN_DIM + j], x[(long)s * XCOLS + 1 + j] * w);
    }
}

// h1 = relu(agg1 @ W1^T + b1)   K=5 — plain VALU, one thread per (node, out)
__global__ void k_gemm1(const float* __restrict__ agg1, const float* __restrict__ W1,
                        const float* __restrict__ b1, float* __restrict__ h1) {
    long t = (long)blockIdx.x * blockDim.x + threadIdx.x;
    if (t < (long)N_NODES * HID) {
        int n = (int)(t >> 6);
        int o = (int)(t & 63);
        float s = b1[o];
#pragma unroll
        for (int k = 0; k < IN_DIM; ++k)
            s += agg1[(long)n * IN_DIM + k] * W1[o * IN_DIM + k];
        h1[t] = fmaxf(s, 0.0f);
    }
}

// ---------------- layer 2: propagate 64-dim h1 ----------------
__global__ void k_self2(const float* __restrict__ h, const float* __restrict__ dis,
                        float* __restrict__ t2) {
    long t = (long)blockIdx.x * blockDim.x + threadIdx.x;
    if (t < (long)N_NODES * HID) {
        int n = (int)(t >> 6);
        float d = dis[n];
        t2[t] = h[t] * d * d;
    }
}

// one wave32 per edge: lane c handles channels c and c+32 (coalesced 128B gathers)
__global__ void k_edge2(const long long* __restrict__ src, const long long* __restrict__ dst,
                        const float* __restrict__ h, const float* __restrict__ dis,
                        float* __restrict__ t2) {
    long tid = (long)blockIdx.x * blockDim.x + threadIdx.x;
    long e = tid >> 5;
    int c = (int)(tid & 31);
    if (e < N_EDGES) {
        int s = (int)src[e], d = (int)dst[e];
        float w = dis[s] * dis[d];
        atomicAdd(&t2[(long)d * HID + c],      h[(long)s * HID + c]      * w);
        atomicAdd(&t2[(long)d * HID + c + 32], h[(long)s * HID + c + 32] * w);
    }
}

// ---------------- layer 2 GEMM: h2 = relu(t2 @ W2^T + b2) via fp32 WMMA ----------------
// block = 128 threads = 4 waves; each wave does a 16-node x 64-col tile.
// A (16x4 f32): lane l -> row m=l&15, holds K = 2*(l>>4) .. +1 within each 4-wide k-step.
// C/D (16x16 f32, 8 VGPRs): lane l, vgpr v -> row v + 8*(l>>4), col l&15.
#define LDSW 66   // padded stride to spread LDS banks
__global__ void __launch_bounds__(128)
k_gemm2(const float* __restrict__ T, const float* __restrict__ W2,
        const float* __restrict__ b2, float* __restrict__ H2) {
    __shared__ float w2s[HID * LDSW];
    int tid = threadIdx.x;
    for (int i = tid; i < HID * HID; i += 128) {
        int r = i >> 6, c = i & 63;
        w2s[r * LDSW + c] = W2[i];          // w2s[col][k] = W2[col*64+k] = (W2^T)[k][col]
    }
    __syncthreads();

    int wave = tid >> 5;
    int lane = tid & 31;
    int m    = lane & 15;                   // A-row / C-col index
    int kh   = lane >> 4;                   // which K half-pair this lane owns
    int base = (blockIdx.x * 4 + wave) * 16;
    if (base >= N_NODES) return;            // wave-uniform: EXEC stays all-1s for WMMA

    v8f a0 = {0,0,0,0,0,0,0,0};
    v8f a1 = {0,0,0,0,0,0,0,0};
    v8f a2 = {0,0,0,0,0,0,0,0};
    v8f a3 = {0,0,0,0,0,0,0,0};

    const float* arow = T + (long)(base + m) * HID;
#pragma unroll
    for (int ks = 0; ks < 16; ++ks) {
        int kb = ks * 4 + kh * 2;                           // 8B-aligned
        v2f a  = *(const v2f*)(arow + kb);                  // A[m][kb..kb+1]
        v2f b0 = *(const v2f*)(&w2s[( 0 + m) * LDSW + kb]); // B[kb..][col]
        v2f b1v = *(const v2f*)(&w2s[(16 + m) * LDSW + kb]);
        v2f b2v = *(const v2f*)(&w2s[(32 + m) * LDSW + kb]);
        v2f b3v = *(const v2f*)(&w2s[(48 + m) * LDSW + kb]);
        a0 = __builtin_amdgcn_wmma_f32_16x16x4_f32(false, a, false, b0,  (short)0, a0, false, false);
        a1 = __builtin_amdgcn_wmma_f32_16x16x4_f32(false, a, false, b1v, (short)0, a1, false, false);
        a2 = __builtin_amdgcn_wmma_f32_16x16x4_f32(false, a, false, b2v, (short)0, a2, false, false);
        a3 = __builtin_amdgcn_wmma_f32_16x16x4_f32(false, a, false, b3v, (short)0, a3, false, false);
    }

    float bi0 = b2[ 0 + m], bi1 = b2[16 + m], bi2 = b2[32 + m], bi3 = b2[48 + m];
#pragma unroll
    for (int v = 0; v < 8; ++v) {
        long row = base + kh * 8 + v;
        float* orow = H2 + row * HID;
        orow[ 0 + m] = fmaxf(a0[v] + bi0, 0.0f);
        orow[16 + m] = fmaxf(a1[v] + bi1, 0.0f);
        orow[32 + m] = fmaxf(a2[v] + bi2, 0.0f);
        orow[48 + m] = fmaxf(a3[v] + bi3, 0.0f);
    }
}

// ---------------- pooling ----------------
__global__ void k_zero_pool(float* pooled, float* cnt) {
    int t = blockIdx.x * blockDim.x + threadIdx.x;
    if (t < N_GRAPHS * HID) pooled[t] = 0.0f;
    if (t < N_GRAPHS) cnt[t] = 0.0f;
}

__global__ void k_pool(const float* __restrict__ H2, const long long* __restrict__ batch,
                       float* __restrict__ pooled, float* __restrict__ cnt) {
    long t = (long)blockIdx.x * blockDim.x + threadIdx.x;
    if (t < (long)N_NODES * HID) {
        int n = (int)(t >> 6);
        int c = (int)(t & 63);
        int g = (int)batch[n];
        atomicAdd(&pooled[g * HID + c], H2[t]);
        if (c == 0) atomicAdd(&cnt[g], 1.0f);
    }
}

// ---------------- head MLP + L2 normalize (tiny: one block, thread per graph) ----------------
__global__ void __launch_bounds__(N_GRAPHS)
k_head(const float* __restrict__ pooled, const float* __restrict__ cnt,
       const float* __restrict__ fc1w, const float* __restrict__ fc1b,
       const float* __restrict__ fc2w, const float* __restrict__ fc2b,
       float* __restrict__ out) {
    __shared__ float sz[N_GRAPHS * HID];    // z activations, 32KB
    int g = threadIdx.x;
    float inv = 1.0f / fmaxf(cnt[g], 1.0f);
    for (int j = 0; j < HID; ++j) {
        float s = fc1b[j];
        for (int k = 0; k < HID; ++k)
            s += fc1w[j * HID + k] * (pooled[g * HID + k] * inv);
        sz[g * HID + j] = fmaxf(s, 0.0f);
    }
    float nrm2 = 0.0f;
    for (int j = 0; j < HID; ++j) {
        float s = fc2b[j];
        for (int k = 0; k < HID; ++k)
            s += fc2w[j * HID + k] * sz[g * HID + k];
        out[g * HID + j] = s;
        nrm2 += s * s;
    }
    float scale = 1.0f / fmaxf(sqrtf(nrm2), 1e-12f);
    for (int j = 0; j < HID; ++j) out[g * HID + j] *= scale;
}

extern "C" void kernel_launch(void* const* d_in, const int* in_sizes, int n_in,
                              void* d_out, int out_size, void* d_ws, size_t ws_size,
                              hipStream_t stream) {
    const float*     x     = (const float*)d_in[0];
    const float*     W1    = (const float*)d_in[1];
    const float*     b1    = (const float*)d_in[2];
    const float*     W2    = (const float*)d_in[3];
    const float*     b2    = (const float*)d_in[4];
    const float*     fc1w  = (const float*)d_in[5];
    const float*     fc1b  = (const float*)d_in[6];
    const float*     fc2w  = (const float*)d_in[7];
    const float*     fc2b  = (const float*)d_in[8];
    const long long* eidx  = (const long long*)d_in[9];
    const long long* batch = (const long long*)d_in[10];
    const long long* src   = eidx;
    const long long* dst   = eidx + N_EDGES;
    float* out = (float*)d_out;
    (void)in_sizes; (void)n_in; (void)out_size; (void)ws_size;

    char*  ws  = (char*)d_ws;
    size_t off = 0;
    auto alloc = [&](size_t bytes) -> void* {
        void* p = ws + off;
        off = (off + bytes + 255) & ~(size_t)255;
        return p;
    };
    unsigned int* deg    = (unsigned int*)alloc((size_t)N_NODES * 4);
    float*        dis    = (float*)alloc((size_t)N_NODES * 4);
    float*        agg1   = (float*)alloc((size_t)N_NODES * IN_DIM * 4);
    float*        bufA   = (float*)alloc((size_t)N_NODES * HID * 4);   // h1, later h2
    float*        bufB   = (float*)alloc((size_t)N_NODES * HID * 4);   // t2 = Ahat@h1
    float*        pooled = (float*)alloc((size_t)N_GRAPHS * HID * 4);
    float*        cnt    = (float*)alloc((size_t)N_GRAPHS * 4);

    const int B = 256;
    int gN   = (N_NODES + B - 1) / B;                       // 391
    int gE   = (N_EDGES + B - 1) / B;                       // 12500
    int gN5  = (N_NODES * IN_DIM + B - 1) / B;              // 1954
    long nh  = (long)N_NODES * HID;
    int gNH  = (int)((nh + B - 1) / B);                     // 25000
    int gE32 = (int)(((long)N_EDGES * 32 + B - 1) / B);     // 400000
    int gT   = (N_NODES + 63) / 64;                         // 1563 (64 nodes / block)

    k_init_deg <<<gN,   B, 0, stream>>>(deg);
    k_count_deg<<<gE,   B, 0, stream>>>(dst, deg);
    k_dis      <<<gN,   B, 0, stream>>>(deg, dis);

    k_self1    <<<gN5,  B, 0, stream>>>(x, dis, agg1);
    k_edge1    <<<gE,   B, 0, stream>>>(src, dst, x, dis, agg1);
    k_gemm1    <<<gNH,  B, 0, stream>>>(agg1, W1, b1, bufA);

    k_self2    <<<gNH,  B, 0, stream>>>(bufA, dis, bufB);
    k_edge2    <<<gE32, B, 0, stream>>>(src, dst, bufA, dis, bufB);
    k_gemm2    <<<gT, 128, 0, stream>>>(bufB, W2, b2, bufA);   // h2 -> bufA

    k_zero_pool<<<(N_GRAPHS * HID + B - 1) / B, B, 0, stream>>>(pooled, cnt);
    k_pool     <<<gNH,  B, 0, stream>>>(bufA, batch, pooled, cnt);
    k_head     <<<1, N_GRAPHS, 0, stream>>>(pooled, cnt, fc1w, fc1b, fc2w, fc2b, out);
}